// get_model_55121610277260
// MI455X (gfx1250) — compile-verified
//
#include <hip/hip_runtime.h>

// =====================================================================
// RandLA-Net style forward pass for gfx1250 (MI455X).
//
// All 1x1 convs / FCs are per-batch GEMMs W(Cout x Cin) * X(Cin x S),
// executed with the native fp32 matrix op V_WMMA_F32_16X16X4_F32:
// one 16x16 output tile per wave32, K-loop in steps of 4.
// BN + bias + LeakyReLU fused into the GEMM epilogue.
//
// Input index assumptions (jax tree-flatten, sorted dict keys):
//   0: points, 1..10: neigh0,interp0,...,neigh4,interp4
//   11..40 : dec[0..4]  (cbr leaves: b,beta,gamma,mean,var,w)
//   41..46 : dec0
//   47..261: enc[0..4], each 43 leaves:
//            +0 ap1.fc_w, +1..6 ap1.mlp, +7 ap2.fc_w, +8..13 ap2.mlp,
//            +14 lfa.mlp1, +20 lfa.mlp2, +26 mlp1, +32 mlp2, +38 shortcut
//   262..267 fc0, 268..273 fc1, 274..279 fc2, 280 fc3.b, 281 fc3.w
// =====================================================================

typedef __attribute__((ext_vector_type(2))) float v2f;
typedef __attribute__((ext_vector_type(8))) float v8f;

#define BB   4
#define NPTS 40960
#define KNN  16

// ---------------------------------------------------------------------
// GEMM + fused epilogue.  Y[b, m, s] = act(bn(sum_c W[m,c] X[b,c,s] + bias))
// mode: 0 = (+bias) linear, 1 = bias+BN+LeakyReLU(0.2), 2 = bias+BN only
// One wave computes one 16x16 tile; 8 waves / block on consecutive n-tiles.
// ---------------------------------------------------------------------
__global__ void __launch_bounds__(256)
gemm_bn_act(const float* __restrict__ W, const float* __restrict__ bias,
            const float* __restrict__ gamma, const float* __restrict__ beta,
            const float* __restrict__ mean,  const float* __restrict__ var,
            const float* __restrict__ X, long xStrideB,
            float* __restrict__ Y, long yStrideB,
            int Cin, int Cout, int S, int mode)
{
    const int lane  = threadIdx.x & 31;
    const int wave  = threadIdx.x >> 5;
    const int ntile = blockIdx.x * 8 + wave;
    const int tilesN = S >> 4;
    if (ntile >= tilesN) return;                 // wave-uniform: EXEC stays full

    const int n0 = ntile << 4;
    const int m0 = blockIdx.y << 4;
    const float* x = X + (long)blockIdx.z * xStrideB + n0;
    float*       y = Y + (long)blockIdx.z * yStrideB + n0;

    const int half = lane >> 4;                  // 0: K={0,1}/M rows 0-7, 1: K={2,3}/M rows 8-15
    const int l    = lane & 15;
    const int kh   = half << 1;
    int arow = m0 + l;                           // A layout: lane -> M row
    if (arow >= Cout) arow = Cout - 1;           // clamp (stores are guarded)
    const float* wr = W + (long)arow * Cin;

    v8f acc = {0.f, 0.f, 0.f, 0.f, 0.f, 0.f, 0.f, 0.f};

    int k = 0;
    for (; k + 4 <= Cin; k += 4) {
        v2f a, b;
        a.x = wr[k + kh];
        a.y = wr[k + kh + 1];
        b.x = x[(long)(k + kh)     * S + l];
        b.y = x[(long)(k + kh + 1) * S + l];
        acc = __builtin_amdgcn_wmma_f32_16x16x4_f32(false, a, false, b,
                                                    (short)0, acc, false, false);
    }
    if (k < Cin) {                               // K remainder (Cin = 6, 10, ...)
        v2f a = {0.f, 0.f}, b = {0.f, 0.f};
        if (k + kh < Cin) {
            a.x = wr[k + kh];
            b.x = x[(long)(k + kh) * S + l];
        }
        if (k + kh + 1 < Cin) {
            a.y = wr[k + kh + 1];
            b.y = x[(long)(k + kh + 1) * S + l];
        }
        acc = __builtin_amdgcn_wmma_f32_16x16x4_f32(false, a, false, b,
                                                    (short)0, acc, false, false);
    }

#pragma unroll
    for (int v = 0; v < 8; ++v) {                // C/D layout: VGPR v -> M = v + 8*half
        const int m = m0 + v + (half << 3);
        if (m < Cout) {
            float val = acc[v];
            if (bias) val += bias[m];
            if (mode >= 1) {
                const float sc = gamma[m] * rsqrtf(var[m] + 1e-5f);
                val = (val - mean[m]) * sc + beta[m];
            }
            if (mode == 1) val = (val > 0.f) ? val : 0.2f * val;
            y[(long)m * S + l] = val;
        }
    }
}

// ---------------------------------------------------------------------
// relative_pos_encoding: out (B,10,N,K) = [dis, rel(3), tile(3), nbr(3)]
// xyz read directly from points (B,6,N0) rows 0..2.
// ---------------------------------------------------------------------
__global__ void __launch_bounds__(256)
relpos_kernel(const float* __restrict__ pts, const int* __restrict__ idx,
              float* __restrict__ out, int N, long total)
{
    long t = (long)blockIdx.x * blockDim.x + threadIdx.x;
    if (t >= total) return;
    const int k = t % KNN;
    const int n = (t / KNN) % N;
    const int b = (int)(t / ((long)KNN * N));
    const int j = idx[((long)b * N + n) * KNN + k];
    const float* P = pts + (long)b * 6 * NPTS;
    const float tx = P[n],            ty = P[NPTS + n],  tz = P[2 * NPTS + n];
    const float nx = P[j],            ny = P[NPTS + j],  nz = P[2 * NPTS + j];
    const float rx = tx - nx, ry = ty - ny, rz = tz - nz;
    const float dis = sqrtf(rx * rx + ry * ry + rz * rz);
    const long cs = (long)N * KNN;
    long o = (long)b * 10 * cs + (long)n * KNN + k;
    out[o] = dis;
    out[o + 1 * cs] = rx; out[o + 2 * cs] = ry; out[o + 3 * cs] = rz;
    out[o + 4 * cs] = tx; out[o + 5 * cs] = ty; out[o + 6 * cs] = tz;
    out[o + 7 * cs] = nx; out[o + 8 * cs] = ny; out[o + 9 * cs] = nz;
}

// gather_neighbour into a channel slice of a (B,Ctot,N,K) concat buffer
__global__ void __launch_bounds__(256)
gather_slice(const float* __restrict__ src, const int* __restrict__ idx,
             float* __restrict__ dst, int C, int Ctot, int coff, int N, long total)
{
    long t = (long)blockIdx.x * blockDim.x + threadIdx.x;
    if (t >= total) return;
    const int k = t % KNN;
    long r = t / KNN;
    const int n = r % N; r /= N;
    const int c = r % C;
    const int b = (int)(r / C);
    const int j = idx[((long)b * N + n) * KNN + k];
    dst[(((long)b * Ctot + coff + c) * N + n) * KNN + k] =
        src[((long)b * C + c) * N + j];
}

// softmax over K + weighted sum: agg[b,c,n] = sum_k f*softmax(att)
__global__ void __launch_bounds__(256)
softmax_agg(const float* __restrict__ f, const float* __restrict__ att,
            float* __restrict__ agg, int C, int N, long total)
{
    long t = (long)blockIdx.x * blockDim.x + threadIdx.x;
    if (t >= total) return;
    const int n = t % N;
    const int c = (t / N) % C;
    const int b = (int)(t / ((long)N * C));
    const long base = (((long)b * C + c) * N + n) * KNN;
    float m = -3.0e38f;
    for (int k = 0; k < KNN; ++k) m = fmaxf(m, att[base + k]);
    float s = 0.f, acc = 0.f;
    for (int k = 0; k < KNN; ++k) {
        const float e = __expf(att[base + k] - m);
        s += e;
        acc += e * f[base + k];
    }
    agg[((long)b * C + c) * N + n] = acc / s;
}

// random_sample: max over K gathered values, rows m < Nout of idx(B,Nin,K)
__global__ void __launch_bounds__(256)
sample_max(const float* __restrict__ f, const int* __restrict__ idx,
           float* __restrict__ out, int C, int Nin, int Nout, long total)
{
    long t = (long)blockIdx.x * blockDim.x + threadIdx.x;
    if (t >= total) return;
    const int m = t % Nout;
    const int c = (t / Nout) % C;
    const int b = (int)(t / ((long)Nout * C));
    const long ib = ((long)b * Nin + m) * KNN;
    const float* fb = f + ((long)b * C + c) * Nin;
    float v = -3.0e38f;
    for (int k = 0; k < KNN; ++k) v = fmaxf(v, fb[idx[ib + k]]);
    out[((long)b * C + c) * Nout + m] = v;
}

// nearest_interp into a channel slice of (B,Ctot,Nout)
__global__ void __launch_bounds__(256)
interp_slice(const float* __restrict__ src, const int* __restrict__ idx,
             float* __restrict__ dst, int C, int Ctot, int coff,
             int Nin, int Nout, long total)
{
    long t = (long)blockIdx.x * blockDim.x + threadIdx.x;
    if (t >= total) return;
    const int n = t % Nout;
    const int c = (t / Nout) % C;
    const int b = (int)(t / ((long)Nout * C));
    const int j = idx[(long)b * Nout + n];
    dst[((long)b * Ctot + coff + c) * Nout + n] =
        src[((long)b * C + c) * Nin + j];
}

// copy (B,C,N) into a channel slice of (B,Ctot,N)
__global__ void __launch_bounds__(256)
copy_slice(const float* __restrict__ src, float* __restrict__ dst,
           int C, int Ctot, int coff, int N, long total)
{
    long t = (long)blockIdx.x * blockDim.x + threadIdx.x;
    if (t >= total) return;
    const int n = t % N;
    const int c = (t / N) % C;
    const int b = (int)(t / ((long)N * C));
    dst[((long)b * Ctot + coff + c) * N + n] = src[((long)b * C + c) * N + n];
}

// residual: out = lrelu(a + b)
__global__ void __launch_bounds__(256)
add_lrelu(const float* __restrict__ a, const float* __restrict__ b,
          float* __restrict__ out, long total)
{
    long t = (long)blockIdx.x * blockDim.x + threadIdx.x;
    if (t >= total) return;
    const float v = a[t] + b[t];
    out[t] = (v > 0.f) ? v : 0.2f * v;
}

// log_softmax over 13 classes, output transposed to (B, N, 13)
__global__ void __launch_bounds__(256)
logsoftmax13(const float* __restrict__ x, float* __restrict__ out, long total)
{
    long t = (long)blockIdx.x * blockDim.x + threadIdx.x;
    if (t >= total) return;
    const int n = t % NPTS;
    const int b = (int)(t / NPTS);
    const float* xb = x + (long)b * 13 * NPTS + n;
    float v[13];
    float m = -3.0e38f;
    for (int c = 0; c < 13; ++c) { v[c] = xb[(long)c * NPTS]; m = fmaxf(m, v[c]); }
    float s = 0.f;
    for (int c = 0; c < 13; ++c) s += __expf(v[c] - m);
    const float lse = m + logf(s);
    float* ob = out + ((long)b * NPTS + n) * 13;
    for (int c = 0; c < 13; ++c) ob[c] = v[c] - lse;
}

// ---------------------------------------------------------------------
// Host side
// ---------------------------------------------------------------------
struct CBR { const float *w, *b, *gamma, *beta, *mean, *var; };

static CBR getCBR(void* const* din, int base) {
    CBR c;
    c.b     = (const float*)din[base + 0];
    c.beta  = (const float*)din[base + 1];
    c.gamma = (const float*)din[base + 2];
    c.mean  = (const float*)din[base + 3];
    c.var   = (const float*)din[base + 4];
    c.w     = (const float*)din[base + 5];
    return c;
}

static void gemmL(hipStream_t st, const float* W, const float* bias, const CBR* bn,
                  int mode, const float* X, long xsb, float* Y, long ysb,
                  int Cin, int Cout, int S)
{
    dim3 grid((unsigned)((S / 16 + 7) / 8), (unsigned)((Cout + 15) / 16), BB);
    gemm_bn_act<<<grid, 256, 0, st>>>(W, bias,
        bn ? bn->gamma : nullptr, bn ? bn->beta : nullptr,
        bn ? bn->mean  : nullptr, bn ? bn->var  : nullptr,
        X, xsb, Y, ysb, Cin, Cout, S, mode);
}

static inline unsigned nblk(long total) { return (unsigned)((total + 255) / 256); }

extern "C" void kernel_launch(void* const* d_in, const int* in_sizes, int n_in,
                              void* d_out, int out_size, void* d_ws, size_t ws_size,
                              hipStream_t stream)
{
    (void)in_sizes; (void)n_in; (void)out_size; (void)ws_size;

    const float* points = (const float*)d_in[0];
    const int* neigh[5], * interp[5];
    for (int i = 0; i < 5; ++i) {
        neigh[i]  = (const int*)d_in[1 + 2 * i];
        interp[i] = (const int*)d_in[2 + 2 * i];
    }

    CBR decp[5];
    for (int j = 0; j < 5; ++j) decp[j] = getCBR(d_in, 11 + 6 * j);
    CBR dec0p = getCBR(d_in, 41);
    struct EncP {
        const float* ap1_fc; CBR ap1_mlp;
        const float* ap2_fc; CBR ap2_mlp;
        CBR lfa_mlp1, lfa_mlp2, mlp1, mlp2, shortcut;
    } encp[5];
    for (int i = 0; i < 5; ++i) {
        const int base = 47 + 43 * i;
        encp[i].ap1_fc   = (const float*)d_in[base + 0];
        encp[i].ap1_mlp  = getCBR(d_in, base + 1);
        encp[i].ap2_fc   = (const float*)d_in[base + 7];
        encp[i].ap2_mlp  = getCBR(d_in, base + 8);
        encp[i].lfa_mlp1 = getCBR(d_in, base + 14);
        encp[i].lfa_mlp2 = getCBR(d_in, base + 20);
        encp[i].mlp1     = getCBR(d_in, base + 26);
        encp[i].mlp2     = getCBR(d_in, base + 32);
        encp[i].shortcut = getCBR(d_in, base + 38);
    }
    CBR fc0 = getCBR(d_in, 262);
    CBR fc1 = getCBR(d_in, 268);
    CBR fc2 = getCBR(d_in, 274);
    const float* fc3_b = (const float*)d_in[280];
    const float* fc3_w = (const float*)d_in[281];

    // ---------------- workspace arena (floats) -----------------------
    float* ws = (float*)d_ws;
    size_t off = 0;
    auto take = [&](size_t n) { float* p = ws + off; off += n; return p; };
    float* F     = take(1310720);    // mlp1 output (B, d/2, N)
    float* CATA  = take(41943040);   // concat 1 (B, d, N, K)
    float* CATB  = take(41943040);   // concat 2 (B, d, N, K)
    float* ATT   = take(41943040);   // attention scores; also rel-pos buffer
    float* AGG   = take(2621440);    // (B, d, N)
    float* FAGG  = take(1310720);    // ap1.mlp out (B, d/2, N)
    float* AP2   = take(2621440);    // ap2.mlp out (B, d, N)
    float* MLP2  = take(5242880);    // cb(mlp2) out (B, 2d, N); also fc3 out
    float* SHORT = take(5242880);    // cb(shortcut) out
    float* FEATA = take(1310720);    // fc0 output (B, 8, N0)
    float* FEATB = take(5242880);    // f_enc temp (i > 0)
    float* ENCS[6];
    const size_t encsz[6] = {5242880, 1310720, 1310720, 655360, 327680, 327680};
    for (int i = 0; i < 6; ++i) ENCS[i] = take(encsz[i]);
    // total ~160M floats (~640 MB)

    const int NS[6] = {40960, 10240, 2560, 640, 160, 80};
    const int DD[5] = {16, 64, 128, 256, 512};
    const int encC[6] = {32, 32, 128, 256, 512, 1024};

    // ---------------- fc0 --------------------------------------------
    gemmL(stream, fc0.w, fc0.b, &fc0, 1, points, (long)6 * NPTS,
          FEATA, (long)8 * NPTS, 6, 8, NPTS);

    const float* feat = FEATA;
    int cin = 8;

    // ---------------- encoder ----------------------------------------
    for (int i = 0; i < 5; ++i) {
        const EncP& E = encp[i];
        const int N = NS[i], d = DD[i], h = d / 2;
        const long NK = (long)N * KNN;

        // f = cbr(mlp1, feat_in)
        gemmL(stream, E.mlp1.w, E.mlp1.b, &E.mlp1, 1,
              feat, (long)cin * N, F, (long)h * N, cin, h, N);

        // rel-pos encoding -> ATT (reused as (B,10,N,K))
        relpos_kernel<<<nblk((long)BB * N * KNN), 256, 0, stream>>>(
            points, neigh[i], ATT, N, (long)BB * N * KNN);

        // f_xyz1 = cbr(lfa.mlp1, relpos) -> CATA slice [h, d)
        gemmL(stream, E.lfa_mlp1.w, E.lfa_mlp1.b, &E.lfa_mlp1, 1,
              ATT, 10 * NK, CATA + h * NK, (long)d * NK, 10, h, (int)NK);

        // f_nb1 = gather(f) -> CATA slice [0, h)
        gather_slice<<<nblk((long)BB * h * NK), 256, 0, stream>>>(
            F, neigh[i], CATA, h, d, 0, N, (long)BB * h * NK);

        // att1 = ap1.fc_w @ cat1 -> ATT
        gemmL(stream, E.ap1_fc, nullptr, nullptr, 0,
              CATA, (long)d * NK, ATT, (long)d * NK, d, d, (int)NK);
        softmax_agg<<<nblk((long)BB * d * N), 256, 0, stream>>>(
            CATA, ATT, AGG, d, N, (long)BB * d * N);

        // f_agg = cbr(ap1.mlp, agg)
        gemmL(stream, E.ap1_mlp.w, E.ap1_mlp.b, &E.ap1_mlp, 1,
              AGG, (long)d * N, FAGG, (long)h * N, d, h, N);

        // f_xyz2 = cbr(lfa.mlp2, f_xyz1) -> CATB slice [h, d)
        gemmL(stream, E.lfa_mlp2.w, E.lfa_mlp2.b, &E.lfa_mlp2, 1,
              CATA + h * NK, (long)d * NK, CATB + h * NK, (long)d * NK,
              h, h, (int)NK);

        // f_nb2 = gather(f_agg) -> CATB slice [0, h)
        gather_slice<<<nblk((long)BB * h * NK), 256, 0, stream>>>(
            FAGG, neigh[i], CATB, h, d, 0, N, (long)BB * h * NK);

        // att2 / agg2 / ap2.mlp
        gemmL(stream, E.ap2_fc, nullptr, nullptr, 0,
              CATB, (long)d * NK, ATT, (long)d * NK, d, d, (int)NK);
        softmax_agg<<<nblk((long)BB * d * N), 256, 0, stream>>>(
            CATB, ATT, AGG, d, N, (long)BB * d * N);
        gemmL(stream, E.ap2_mlp.w, E.ap2_mlp.b, &E.ap2_mlp, 1,
              AGG, (long)d * N, AP2, (long)d * N, d, d, N);

        // mlp2 (cb) + shortcut (cb) + residual lrelu
        gemmL(stream, E.mlp2.w, E.mlp2.b, &E.mlp2, 2,
              AP2, (long)d * N, MLP2, (long)2 * d * N, d, 2 * d, N);
        gemmL(stream, E.shortcut.w, E.shortcut.b, &E.shortcut, 2,
              feat, (long)cin * N, SHORT, (long)2 * d * N, cin, 2 * d, N);
        float* fenc = (i == 0) ? ENCS[0] : FEATB;
        add_lrelu<<<nblk((long)BB * 2 * d * N), 256, 0, stream>>>(
            MLP2, SHORT, fenc, (long)BB * 2 * d * N);

        // random_sample -> ENCS[i+1]
        sample_max<<<nblk((long)BB * 2 * d * NS[i + 1]), 256, 0, stream>>>(
            fenc, neigh[i], ENCS[i + 1], 2 * d, N, NS[i + 1],
            (long)BB * 2 * d * NS[i + 1]);

        feat = ENCS[i + 1];
        cin = 2 * d;
    }

    // ---------------- decoder ----------------------------------------
    gemmL(stream, dec0p.w, dec0p.b, &dec0p, 1,
          ENCS[5], (long)1024 * 80, CATB, (long)1024 * 80, 1024, 1024, 80);
    const float* dfeat = CATB;
    int dc = 1024;
    const int decOutC[5] = {512, 256, 128, 32, 32};

    for (int j = 0; j < 5; ++j) {
        const int lvl  = 4 - j;
        const int Nin  = NS[lvl + 1];
        const int Nout = NS[lvl];
        const float* skip = ENCS[lvl];
        const int skipC = encC[lvl];
        const int Ctot  = skipC + dc;

        copy_slice<<<nblk((long)BB * skipC * Nout), 256, 0, stream>>>(
            skip, CATA, skipC, Ctot, 0, Nout, (long)BB * skipC * Nout);
        interp_slice<<<nblk((long)BB * dc * Nout), 256, 0, stream>>>(
            dfeat, interp[lvl], CATA, dc, Ctot, skipC, Nin, Nout,
            (long)BB * dc * Nout);

        float* outbuf = (dfeat == CATB) ? ATT : CATB;
        gemmL(stream, decp[j].w, decp[j].b, &decp[j], 1,
              CATA, (long)Ctot * Nout, outbuf, (long)decOutC[j] * Nout,
              Ctot, decOutC[j], Nout);
        dfeat = outbuf;
        dc = decOutC[j];
    }

    // ---------------- head -------------------------------------------
    float* t1 = (dfeat == CATB) ? ATT : CATB;
    gemmL(stream, fc1.w, fc1.b, &fc1, 1,
          dfeat, (long)32 * NPTS, t1, (long)64 * NPTS, 32, 64, NPTS);
    float* t2 = (t1 == CATB) ? ATT : CATB;
    gemmL(stream, fc2.w, fc2.b, &fc2, 1,
          t1, (long)64 * NPTS, t2, (long)32 * NPTS, 64, 32, NPTS);
    gemmL(stream, fc3_w, fc3_b, nullptr, 0,
          t2, (long)32 * NPTS, MLP2, (long)13 * NPTS, 32, 13, NPTS);
    logsoftmax13<<<nblk((long)BB * NPTS), 256, 0, stream>>>(
        MLP2, (float*)d_out, (long)BB * NPTS);
}